// Position_attention_22677427322969
// MI455X (gfx1250) — compile-verified
//
#include <hip/hip_runtime.h>
#include <hip/hip_bf16.h>

// MI455X / gfx1250 fused position attention.
// Phase 1: q/k/v projections via V_WMMA_F32_16X16X4_F32 (f32 WMMA keeps
//          reference precision; total problem is only ~19 GFLOP).
// Phase 2: flash-attention style fused energy -> softmax -> attn@V so the
//          268 MB energy tensor never touches HBM; q/k/v (10 MB) stay in L2.
//          Softmax is full-wave parallel via ds_swizzle XOR reductions that
//          match the WMMA C-fragment row grouping (16-lane halves).

typedef float v2f __attribute__((ext_vector_type(2)));
typedef float v8f __attribute__((ext_vector_type(8)));

union F8 { v8f v; float f[8]; };

#define WMMA_F32x4(a, b, c) \
  __builtin_amdgcn_wmma_f32_16x16x4_f32(false, (a), false, (b), (short)0, (c), false, false)

// XOR-lane swizzle within the wave (group-of-32 mode: offset[14:10]=xor,
// [9:5]=or, [4:0]=and). Masks < 16 never cross the 16-lane halves, which is
// exactly the row partition of the 16x16 f32 C/D fragment layout.
template <int MASK>
__device__ __forceinline__ float swz_xor(float v) {
  int i = __builtin_bit_cast(int, v);
  i = __builtin_amdgcn_ds_swizzle(i, (MASK << 10) | 0x1f);
  return __builtin_bit_cast(float, i);
}

__device__ __forceinline__ float redmax16(float v) {
  v = fmaxf(v, swz_xor<8>(v));
  v = fmaxf(v, swz_xor<4>(v));
  v = fmaxf(v, swz_xor<2>(v));
  v = fmaxf(v, swz_xor<1>(v));
  return v;
}

__device__ __forceinline__ float redsum16(float v) {
  v += swz_xor<8>(v);
  v += swz_xor<4>(v);
  v += swz_xor<2>(v);
  v += swz_xor<1>(v);
  return v;
}

// ---------------------------------------------------------------------------
// Kernel 1: per-voxel channel projections.
//   xf [R=B*N, C=128] -> q [R,16], k [R,16], v [R,128]
// One wave (32 lanes) per 16 rows. A-frag shared across all 10 output tiles.
// ---------------------------------------------------------------------------
__global__ __launch_bounds__(32) void qkv_proj_kernel(
    const float* __restrict__ x,
    const float* __restrict__ Wq, const float* __restrict__ Wk,
    const float* __restrict__ Wv,
    float* __restrict__ q, float* __restrict__ k, float* __restrict__ v)
{
  constexpr int C = 128, CQ = 16;
  const int lane = threadIdx.x;
  const int half = lane >> 4;    // 0: lanes 0-15, 1: lanes 16-31
  const int l16  = lane & 15;
  const long long i0 = (long long)blockIdx.x * 16;   // row base in [0, B*N)
  const float* __restrict__ xrow = x + (i0 + l16) * C;

  F8 accq = {}; F8 acck = {}; F8 accv[8] = {};

  for (int kk = 0; kk < 32; ++kk) {          // K loop over C in steps of 4
    const int ka = kk * 4 + half * 2;        // this lane's K indices: ka, ka+1
    v2f a; a.x = xrow[ka]; a.y = xrow[ka + 1];

    v2f bq; bq.x = Wq[ka * CQ + l16]; bq.y = Wq[(ka + 1) * CQ + l16];
    v2f bk; bk.x = Wk[ka * CQ + l16]; bk.y = Wk[(ka + 1) * CQ + l16];
    accq.v = WMMA_F32x4(a, bq, accq.v);
    acck.v = WMMA_F32x4(a, bk, acck.v);

#pragma unroll
    for (int t = 0; t < 8; ++t) {            // 8 column tiles of Wv
      v2f bv; bv.x = Wv[ka * C + t * 16 + l16];
      bv.y        = Wv[(ka + 1) * C + t * 16 + l16];
      accv[t].v = WMMA_F32x4(a, bv, accv[t].v);
    }
  }

  // C/D layout: VGPR r, lane -> row = r + 8*half, col = l16
#pragma unroll
  for (int r = 0; r < 8; ++r) {
    const long long row = i0 + r + 8 * half;
    q[row * CQ + l16] = accq.f[r];
    k[row * CQ + l16] = acck.f[r];
#pragma unroll
    for (int t = 0; t < 8; ++t)
      v[row * C + t * 16 + l16] = accv[t].f[r];
  }
}

// ---------------------------------------------------------------------------
// Kernel 2: fused attention.
//   S[i,j] = k_i . q_j ; P = softmax_j(S) ; out_i = sum_j P[i,j] v_j
//   result = gamma * out + x
// One wave owns 16 rows i; streams j in tiles of 16 with online softmax.
// O accumulator: 16x128 f32 = 8 C-tiles (64 VGPRs/lane).
// Per-row state (m, l) replicated across each 16-lane half for the 8 rows the
// lane touches in the C-fragment layout; no LDS broadcast needed.
// ---------------------------------------------------------------------------
__global__ __launch_bounds__(32) void attn_kernel(
    const float* __restrict__ q, const float* __restrict__ k,
    const float* __restrict__ v, const float* __restrict__ x,
    const float* __restrict__ gamma_p, float* __restrict__ out, int N)
{
  constexpr int C = 128, CQ = 16;
  __shared__ float St[16][17];      // P tile transpose buffer (padded)

  const int lane = threadIdx.x;
  const int half = lane >> 4;
  const int l16  = lane & 15;
  const int tilesPerBatch = N / 16;                 // 256
  const int b    = blockIdx.x / tilesPerBatch;
  const int tile = blockIdx.x % tilesPerBatch;
  const long long base = (long long)b * N;          // batch row base
  const long long i0   = base + (long long)tile * 16;

  // Per-row online-softmax state for rows r + 8*half (replicated per lane).
  float m_state[8], l_state[8];
#pragma unroll
  for (int r = 0; r < 8; ++r) { m_state[r] = -3.402823e38f; l_state[r] = 0.0f; }

  F8 o[8] = {};

  // Preload this block's K rows as A-fragments (K-dim = CQ = 16 -> 4 steps).
  float afr[8];
#pragma unroll
  for (int kk = 0; kk < 4; ++kk) {
    const int ka = kk * 4 + half * 2;
    afr[kk * 2]     = k[(i0 + l16) * CQ + ka];
    afr[kk * 2 + 1] = k[(i0 + l16) * CQ + ka + 1];
  }

  for (int j0 = 0; j0 < N; j0 += 16) {
    const long long jbase = base + j0;

    // ---- S tile: A = K rows (16 x 16), B = Q^T (16 x 16) -----------------
    v8f s = {};
#pragma unroll
    for (int kk = 0; kk < 4; ++kk) {
      const int ka = kk * 4 + half * 2;
      v2f a; a.x = afr[kk * 2]; a.y = afr[kk * 2 + 1];
      v2f bq; bq.x = q[(jbase + l16) * CQ + ka];
      bq.y        = q[(jbase + l16) * CQ + ka + 1];
      s = WMMA_F32x4(a, bq, s);
    }

    // ---- full-wave online softmax in C-layout ----------------------------
    // Row r+8*half: its 16 columns live in the lane's 16-lane half.
    F8 sf; sf.v = s;
    float p[8], sc[8];
#pragma unroll
    for (int r = 0; r < 8; ++r) {
      const float mt    = redmax16(sf.f[r]);
      const float m_new = fmaxf(m_state[r], mt);
      sc[r] = __expf(m_state[r] - m_new);
      m_state[r] = m_new;
      p[r] = __expf(sf.f[r] - m_new);
      const float ts = redsum16(p[r]);
      l_state[r] = l_state[r] * sc[r] + ts;
    }

    // P: C-layout -> row-major LDS (for A-layout re-read)
#pragma unroll
    for (int r = 0; r < 8; ++r) St[r + 8 * half][l16] = p[r];
    __syncthreads();

    // ---- rescale accumulator rows ---------------------------------------
#pragma unroll
    for (int t = 0; t < 8; ++t)
#pragma unroll
      for (int r = 0; r < 8; ++r) o[t].f[r] *= sc[r];

    // ---- O += P (16x16) @ V tile (16x128) -------------------------------
#pragma unroll
    for (int kk = 0; kk < 4; ++kk) {
      const int ka = kk * 4 + half * 2;
      v2f a; a.x = St[l16][ka]; a.y = St[l16][ka + 1];   // P in A-layout
#pragma unroll
      for (int t = 0; t < 8; ++t) {
        v2f bv; bv.x = v[(jbase + ka) * C + t * 16 + l16];
        bv.y        = v[(jbase + ka + 1) * C + t * 16 + l16];
        o[t].v = WMMA_F32x4(a, bv, o[t].v);
      }
    }
    __syncthreads();   // St reused next iteration
  }

  // ---- epilogue: out = gamma * (O / l) + x -------------------------------
  const float g = *gamma_p;
#pragma unroll
  for (int r = 0; r < 8; ++r) {
    const float inv = 1.0f / l_state[r];
    const long long row = i0 + r + 8 * half;
#pragma unroll
    for (int t = 0; t < 8; ++t) {
      const long long idx = row * C + t * 16 + l16;
      out[idx] = g * (o[t].f[r] * inv) + x[idx];
    }
  }
}

// ---------------------------------------------------------------------------
extern "C" void kernel_launch(void* const* d_in, const int* in_sizes, int n_in,
                              void* d_out, int out_size, void* d_ws, size_t ws_size,
                              hipStream_t stream) {
  const float* x     = (const float*)d_in[0];
  const float* Wq    = (const float*)d_in[1];
  const float* Wk    = (const float*)d_in[2];
  const float* Wv    = (const float*)d_in[3];
  const float* gamma = (const float*)d_in[4];
  float* out = (float*)d_out;
  float* ws  = (float*)d_ws;

  constexpr int B = 4, N = 4096, CQ = 16;
  const long long R = (long long)B * N;              // 16384 rows

  // Workspace layout (floats): q [R*16] | k [R*16] | v [R*128]  -> 10 MB
  float* q = ws;
  float* k = ws + R * CQ;
  float* v = ws + 2 * R * CQ;

  const int rowBlocks = (int)(R / 16);               // 1024
  qkv_proj_kernel<<<rowBlocks, 32, 0, stream>>>(x, Wq, Wk, Wv, q, k, v);
  attn_kernel<<<rowBlocks, 32, 0, stream>>>(q, k, v, x, gamma, out, N);
}